// SphericalGraphCNN_11373073400409
// MI455X (gfx1250) — compile-verified
//
#include <hip/hip_runtime.h>
#include <hip/hip_bf16.h>

typedef __attribute__((ext_vector_type(2))) float v2f;
typedef __attribute__((ext_vector_type(8))) float v8f;
typedef __attribute__((ext_vector_type(4))) int   i32x4;
typedef __attribute__((ext_vector_type(8))) int   i32x8;

#define BATCH 64
#define BN_EPS 1e-5f

// -------------------------------------------------------------------------
// pack: x [B, V] (Fin=1) -> T0 [V, B]
// -------------------------------------------------------------------------
__global__ void pack_x_kernel(const float* __restrict__ x, float* __restrict__ T0, int V) {
  int tid = blockIdx.x * blockDim.x + threadIdx.x;
  int total = V * BATCH;
  if (tid >= total) return;
  int v = tid / BATCH;
  int b = tid - v * BATCH;
  T0[tid] = x[b * V + v];          // x[b, v, 0]
}

// -------------------------------------------------------------------------
// Structured SPMM. Graph COO is 9 stacked arange-blocks: row v's nonzero t
// lives at index t*V+v. Out[v,m] = alpha * sum_t vals[tV+v]*X[cols[tV+v], m]
//                                 + beta * Prev[v,m]
// -------------------------------------------------------------------------
__global__ void spmm_kernel(const int* __restrict__ cols, const float* __restrict__ vals,
                            const float* __restrict__ X, const float* __restrict__ Prev,
                            float* __restrict__ Out, int V, int M, float alpha, float beta) {
  long tid = (long)blockIdx.x * blockDim.x + threadIdx.x;
  long total = (long)V * M;
  if (tid >= total) return;
  int v = (int)(tid / M);
  int m = (int)(tid - (long)v * M);
  float acc = 0.f;
#pragma unroll
  for (int t = 0; t < 9; ++t) {
    int   c = cols[t * V + v];
    float w = vals[t * V + v];
    acc = fmaf(w, X[(long)c * M + m], acc);
  }
  float out = alpha * acc;
  if (beta != 0.f) out = fmaf(beta, Prev[tid], out);
  Out[tid] = out;
}

// -------------------------------------------------------------------------
// Fused ChebConv GEMM + BN + ReLU + 4:1 max-pool + next-layer transpose.
// Block = 8 waves sharing one 16-column W panel staged in LDS; big panels
// are DMA'd by the Tensor Data Mover (tensor_load_to_lds, TENSORcnt),
// small ones cooperatively loaded. One wave per 16x16 D tile,
// V_WMMA_F32_16X16X4_F32 over Kd = Fin*4 (one k-step = 4 Chebyshev orders
// of one input feature).
// -------------------------------------------------------------------------
__global__ void cheb_gemm_fused(const float* __restrict__ T0, const float* __restrict__ T1,
                                const float* __restrict__ T2, const float* __restrict__ T3,
                                const float* __restrict__ W,  const float* __restrict__ gamma,
                                const float* __restrict__ beta, const float* __restrict__ mean,
                                const float* __restrict__ var, float* __restrict__ OutNext,
                                int V, int Fin, int Fout, int mBlocks) {
  __shared__ float Wlds[16384];               // up to (1024 x 16) fp32 panel = 64 KB
  const int Kd   = Fin * 4;
  const int bm   = blockIdx.x % mBlocks;
  const int tn   = blockIdx.x / mBlocks;
  const int col0 = tn << 4;
  const int wid  = threadIdx.x >> 5;
  const int lane = threadIdx.x & 31;

  // ---- stage W[:, col0:col0+16] into LDS -------------------------------
  if (Kd >= 64) {
    if (threadIdx.x < 32) {                   // single wave issues the TDM op
      unsigned long long ga = (unsigned long long)(const void*)(W + col0);
      i32x4 g0; i32x8 g1;
      // D# group0: count=1 | lds_addr=0 | global_addr | type=2 ("image")
      g0[0] = 1;
      g0[1] = 0;
      g0[2] = (int)(unsigned)ga;
      g0[3] = (int)(((unsigned)(ga >> 32) & 0x01FFFFFFu) | (2u << 30));
      // D# group1: data_size=4B; tensor_dim0=Fout, tensor_dim1=Kd;
      //            tile_dim0=16, tile_dim1=Kd; tensor_dim0_stride=Fout
      g1[0] = 2 << 16;                        // data_size = 2 -> 4 bytes
      g1[1] = (Fout & 0xFFFF) << 16;          // tensor_dim0[15:0]
      g1[2] = (Kd & 0xFFFF) << 16;            // tensor_dim0[31:16]=0 | tensor_dim1[15:0]
      g1[3] = 16 << 16;                       // tensor_dim1[31:16]=0 | tile_dim0=16
      g1[4] = Kd & 0xFFFF;                    // tile_dim1=Kd | tile_dim2=0
      g1[5] = Fout;                           // tensor_dim0_stride[31:0]
      g1[6] = 0;                              // stride hi | tensor_dim1_stride lo
      g1[7] = 0;
      asm volatile("tensor_load_to_lds %0, %1" :: "s"(g0), "s"(g1) : "memory");
      __builtin_amdgcn_s_wait_tensorcnt(0);
    }
  } else {                                    // tiny panel: plain cooperative load
    for (int idx = threadIdx.x; idx < Kd * 16; idx += 256)
      Wlds[idx] = W[(idx >> 4) * Fout + col0 + (idx & 15)];
  }
  __syncthreads();

  // ---- WMMA main loop ---------------------------------------------------
  const int lhalf = lane >> 4;                // 0: K=0,1   1: K=2,3
  const int l16   = lane & 15;
  const int tm    = bm * 8 + wid;
  const int row0  = tm << 4;
  const int r     = row0 + l16;               // global A row -> (b, v)
  const int b     = r / V;
  const int v     = r - b * V;
  const long aoff = (long)v * (BATCH * Fin) + (long)b * Fin;
  const float* Ta = (lhalf ? T2 : T0) + aoff; // A VGPR0: K=0 | K=2
  const float* Tb = (lhalf ? T3 : T1) + aoff; // A VGPR1: K=1 | K=3
  const float* wl = Wlds + 32 * lhalf + l16;  // B rows 2*lhalf(+1), col l16

  v8f acc = {0.f, 0.f, 0.f, 0.f, 0.f, 0.f, 0.f, 0.f};
  for (int f = 0; f < Fin; ++f) {
    v2f a, bb;
    a.x  = Ta[f];
    a.y  = Tb[f];
    bb.x = wl[0];
    bb.y = wl[16];
    wl  += 64;
    acc = __builtin_amdgcn_wmma_f32_16x16x4_f32(false, a, false, bb,
                                                (short)0, acc, false, false);
  }

  // ---- BN + ReLU + max-pool(4) + transpose into next T0 [V/4, B*Fout] ---
  const int   o  = col0 + l16;
  const float s  = gamma[o] * rsqrtf(var[o] + BN_EPS);
  const float bt = beta[o];
  const float mu = mean[o];
  const int   Vp = V >> 2;
  const long  strideNext = (long)BATCH * Fout;
#pragma unroll
  for (int pp = 0; pp < 2; ++pp) {
    float e0 = fmaxf(fmaf(s, acc[4 * pp + 0] - mu, bt), 0.f);
    float e1 = fmaxf(fmaf(s, acc[4 * pp + 1] - mu, bt), 0.f);
    float e2 = fmaxf(fmaf(s, acc[4 * pp + 2] - mu, bt), 0.f);
    float e3 = fmaxf(fmaf(s, acc[4 * pp + 3] - mu, bt), 0.f);
    float p  = fmaxf(fmaxf(e0, e1), fmaxf(e2, e3));
    int rp   = (row0 >> 2) + 2 * lhalf + pp;  // pooled row = b*Vp + vp
    int b2   = rp / Vp;
    int vp   = rp - b2 * Vp;
    OutNext[(long)vp * strideNext + (long)b2 * Fout + o] = p;
  }
}

// -------------------------------------------------------------------------
// Small FC + ReLU:  Y[b, j] = relu(X[b,:] @ W[:, j] + bias[j])
// -------------------------------------------------------------------------
__global__ void fc_relu_kernel(const float* __restrict__ X, const float* __restrict__ W,
                               const float* __restrict__ bias, float* __restrict__ Y,
                               int Kdim, int N) {
  int tid = blockIdx.x * blockDim.x + threadIdx.x;
  if (tid >= BATCH * N) return;
  int b = tid / N;
  int j = tid - b * N;
  const float* xr = X + b * Kdim;
  float acc = bias[j];
  for (int i = 0; i < Kdim; ++i) acc = fmaf(xr[i], W[(long)i * N + j], acc);
  Y[tid] = fmaxf(acc, 0.f);
}

// -------------------------------------------------------------------------
extern "C" void kernel_launch(void* const* d_in, const int* in_sizes, int n_in,
                              void* d_out, int out_size, void* d_ws, size_t ws_size,
                              hipStream_t stream) {
  // setup_inputs order: x, weights[7], gammas[7], betas[7], bn_means[7],
  // bn_vars[7], fc1_w, fc1_b, fc2_w, fc2_b, rows[7], cols[7], vals[7]
  const float* x = (const float*)d_in[0];
  const float *Wt[7], *Gm[7], *Bt[7], *Mu[7], *Vr[7], *Vl[7];
  const int* Cl[7];
  for (int i = 0; i < 7; ++i) {
    Wt[i] = (const float*)d_in[1 + i];
    Gm[i] = (const float*)d_in[8 + i];
    Bt[i] = (const float*)d_in[15 + i];
    Mu[i] = (const float*)d_in[22 + i];
    Vr[i] = (const float*)d_in[29 + i];
    Cl[i] = (const int*)  d_in[47 + i];
    Vl[i] = (const float*)d_in[54 + i];
  }
  const float* fc1w = (const float*)d_in[36];
  const float* fc1b = (const float*)d_in[37];
  const float* fc2w = (const float*)d_in[38];
  const float* fc2b = (const float*)d_in[39];

  const int Vs[7] = {16384, 4096, 1024, 256, 64, 16, 4};
  const int Fi[7] = {1, 32, 64, 128, 256, 256, 256};
  const int Fo[7] = {32, 64, 128, 256, 256, 256, 256};

  float* ws = (float*)d_ws;
  const long TCAP = 8388608;                 // max V*B*Fin (layer 1)
  float* slot0 = ws;
  float* slot1 = ws + TCAP;
  float* slot2 = ws + 2 * TCAP;
  float* slot3 = ws + 3 * TCAP;
  float* slot4 = ws + 4 * TCAP;
  float* fc1out = ws + 5 * TCAP;             // 64*512 floats

  // Layer 0 input transpose: x [B,16384,1] -> T0 [16384, 64]
  {
    int total = 16384 * BATCH;
    pack_x_kernel<<<(total + 255) / 256, 256, 0, stream>>>(x, slot0, 16384);
  }

  for (int i = 0; i < 7; ++i) {
    int V = Vs[i], Fin = Fi[i], Fout = Fo[i];
    int M = BATCH * Fin;
    float* T0    = (i & 1) ? slot4 : slot0;
    float* Tnext = (i & 1) ? slot0 : slot4;  // next layer's T0 (pooled+transposed)
    long totalVM = (long)V * M;
    int blocks = (int)((totalVM + 255) / 256);
    // Chebyshev recurrence: T1 = L x ; T2 = 2 L T1 - T0 ; T3 = 2 L T2 - T1
    spmm_kernel<<<blocks, 256, 0, stream>>>(Cl[i], Vl[i], T0, T0, slot1, V, M, 1.f, 0.f);
    spmm_kernel<<<blocks, 256, 0, stream>>>(Cl[i], Vl[i], slot1, T0, slot2, V, M, 2.f, -1.f);
    spmm_kernel<<<blocks, 256, 0, stream>>>(Cl[i], Vl[i], slot2, slot1, slot3, V, M, 2.f, -1.f);

    int tilesM  = (BATCH * V) / 16;          // multiple of 8 for all layers
    int tilesN  = Fout / 16;
    int mBlocks = tilesM / 8;                // 8 row-tiles (waves) per block
    int gblocks = mBlocks * tilesN;
    cheb_gemm_fused<<<gblocks, 256, 0, stream>>>(T0, slot1, slot2, slot3, Wt[i],
                                                 Gm[i], Bt[i], Mu[i], Vr[i],
                                                 Tnext, V, Fin, Fout, mBlocks);
  }

  // Layer 6 (even) wrote slot4 with Vp=1 -> layout is exactly [B=64, 256]
  fc_relu_kernel<<<(BATCH * 512 + 255) / 256, 256, 0, stream>>>(slot4, fc1w, fc1b, fc1out, 256, 512);
  fc_relu_kernel<<<(BATCH * 64  + 255) / 256, 256, 0, stream>>>(fc1out, fc2w, fc2b, (float*)d_out, 512, 64);
}